// Net_18184891531554
// MI455X (gfx1250) — compile-verified
//
#include <hip/hip_runtime.h>

typedef __attribute__((ext_vector_type(2))) float v2f;
typedef __attribute__((ext_vector_type(8))) float v8f;

#define D      128
#define BBLK   5
#define BD     640      // B * D
#define GG     512      // graphs
#define CC     10       // classes
#define EPSG   128.0f   // GIN eps
#define BNEPS  1e-5f

// ---------------- elementwise: agg = (1+eps) * x ----------------
__global__ __launch_bounds__(256) void k_scale(const float* __restrict__ x,
                                               float* __restrict__ y, long n) {
  long t = (long)blockIdx.x * blockDim.x + threadIdx.x;
  if (t < n) y[t] = (1.0f + EPSG) * x[t];
}

// ---------------- edge scatter: agg[dst] += x[src] ----------------
// one wave per edge, 4 floats per lane; x and agg are L2-resident (25.6MB each
// vs 192MB L2) so this runs at L2 atomic throughput, not HBM.
__global__ __launch_bounds__(256) void k_edge_scatter(const float* __restrict__ x,
                                                      const int* __restrict__ src,
                                                      const int* __restrict__ dst,
                                                      float* __restrict__ agg, int E) {
  long t = (long)blockIdx.x * blockDim.x + threadIdx.x;
  int e = (int)(t >> 5);
  if (e >= E) return;
  int lane = (int)(t & 31);
  int s = src[e], d = dst[e];
  const float4 v = ((const float4*)(x + (long)s * D))[lane];
  float* a = agg + (long)d * D + lane * 4;
  atomicAdd(a + 0, v.x);
  atomicAdd(a + 1, v.y);
  atomicAdd(a + 2, v.z);
  atomicAdd(a + 3, v.w);
}

// ---------------- GEMM + bias + relu via V_WMMA_F32_16X16X4_F32 ----------------
// Y[N,128] = relu(X[N,128] @ W[128,128] + bias). One wave per 16-row strip,
// 8 col-tiles of 16x16 f32 accumulators, K stepped by 4. W staged transposed
// in LDS (64KB) so B fragments are contiguous ds_load_b64.
__global__ __launch_bounds__(256) void k_gemm_relu(const float* __restrict__ X,
                                                   const float* __restrict__ W,
                                                   const float* __restrict__ bias,
                                                   float* __restrict__ Y, int nrows) {
  __shared__ float Wt[D * D];   // Wt[c*128 + k] = W[k*128 + c]
  for (int idx = threadIdx.x; idx < D * D; idx += 256) {
    int k = idx >> 7, c = idx & 127;
    Wt[c * D + k] = W[idx];
  }
  __syncthreads();

  int wave = threadIdx.x >> 5;
  int lane = threadIdx.x & 31;
  int strip = blockIdx.x * 8 + wave;        // 16-row strip index
  int nStrips = nrows >> 4;                 // N divisible by 16
  if (strip >= nStrips) return;             // wave-uniform exit (EXEC all-1 for WMMA)

  int halfSel = lane >> 4;                  // 0: K pair {0,1}; 1: K pair {2,3}
  int l15 = lane & 15;
  int kOff = halfSel * 2;
  const float* Xrow = X + (long)(strip * 16 + l15) * D;

  v8f acc[8] = {};
  for (int k0 = 0; k0 < D; k0 += 4) {
    v2f a;
    a.x = Xrow[k0 + kOff];
    a.y = Xrow[k0 + kOff + 1];
#pragma unroll
    for (int j = 0; j < 8; j++) {
      const float* wc = &Wt[(j * 16 + l15) * D + k0 + kOff];
      v2f b;
      b.x = wc[0];
      b.y = wc[1];
      acc[j] = __builtin_amdgcn_wmma_f32_16x16x4_f32(false, a, false, b,
                                                     (short)0, acc[j], false, false);
    }
  }

  // C/D layout: lanes 0-15 -> M = 0..7 over VGPR0..7; lanes 16-31 -> M = 8..15
  int rowBase = strip * 16 + (halfSel ? 8 : 0);
#pragma unroll
  for (int j = 0; j < 8; j++) {
    int col = j * 16 + l15;
    float bv = bias[col];
#pragma unroll
    for (int r = 0; r < 8; r++) {
      float v = acc[j][r] + bv;
      Y[(long)(rowBase + r) * D + col] = v > 0.f ? v : 0.f;
    }
  }
}

// ---------------- column sum / sumsq over [nrows,128] ----------------
__global__ __launch_bounds__(256) void k_colstats128(const float* __restrict__ X,
                                                     int nrows, float* __restrict__ sums) {
  int col = threadIdx.x & 127;
  int half = threadIdx.x >> 7;
  float s = 0.f, q = 0.f;
  for (int r = blockIdx.x * 2 + half; r < nrows; r += gridDim.x * 2) {
    float v = X[(long)r * D + col];
    s += v;
    q += v * v;
  }
  __shared__ float red[512];
  red[threadIdx.x] = s;
  red[256 + threadIdx.x] = q;
  __syncthreads();
  if (half == 0) {
    atomicAdd(&sums[col], s + red[128 + col]);
    atomicAdd(&sums[D + col], q + red[256 + 128 + col]);
  }
}

// ---------------- generic small column stats (classifier head) ----------------
__global__ __launch_bounds__(256) void k_colstats_small(const float* __restrict__ X,
                                                        int nrows, int ncols,
                                                        float* __restrict__ sums) {
  int col = blockIdx.x * blockDim.x + threadIdx.x;
  if (col >= ncols) return;
  float s = 0.f, q = 0.f;
  for (int r = 0; r < nrows; r++) {
    float v = X[(long)r * ncols + col];
    s += v;
    q += v * v;
  }
  sums[col] = s;
  sums[ncols + col] = q;
}

// ---------------- BatchNorm apply (training-mode, biased var) ----------------
__global__ __launch_bounds__(256) void k_bn(const float* __restrict__ X,
                                            float* __restrict__ Y,
                                            const float* __restrict__ sums,
                                            const float* __restrict__ gamma,
                                            const float* __restrict__ beta,
                                            int nrows, int ncols) {
  long t = (long)blockIdx.x * blockDim.x + threadIdx.x;
  long total = (long)nrows * ncols;
  if (t >= total) return;
  int col = (int)(t % ncols);
  float inv = 1.0f / (float)nrows;
  float mu = sums[col] * inv;
  float var = sums[ncols + col] * inv - mu * mu;
  float rs = rsqrtf(var + BNEPS);
  Y[t] = gamma[col] * (X[t] - mu) * rs + beta[col];
}

// ---------------- global add pool: feats[batch[n], colOff:colOff+128] += x[n] ----------------
__global__ __launch_bounds__(256) void k_pool(const float* __restrict__ x,
                                              const int* __restrict__ batch,
                                              float* __restrict__ feats,
                                              int nrows, int colOff) {
  long t = (long)blockIdx.x * blockDim.x + threadIdx.x;
  int n = (int)(t >> 5);
  if (n >= nrows) return;
  int lane = (int)(t & 31);
  int g = batch[n];
  const float4 v = ((const float4*)(x + (long)n * D))[lane];
  float* f = feats + (long)g * BD + colOff + lane * 4;
  atomicAdd(f + 0, v.x);
  atomicAdd(f + 1, v.y);
  atomicAdd(f + 2, v.z);
  atomicAdd(f + 3, v.w);
}

// ---------------- classifier fc1: [G,640] @ [640,128] + bias, relu ----------------
__global__ __launch_bounds__(256) void k_fc1(const float* __restrict__ fbn,
                                             const float* __restrict__ Wc1,
                                             const float* __restrict__ bc1,
                                             float* __restrict__ out) {
  int t = blockIdx.x * blockDim.x + threadIdx.x;   // G*128
  if (t >= GG * D) return;
  int g = t >> 7, d = t & 127;
  float s = bc1[d];
  const float* row = fbn + (long)g * BD;
  for (int k = 0; k < BD; k++) s += row[k] * Wc1[k * D + d];
  out[t] = s > 0.f ? s : 0.f;
}

// ---------------- fc2 + log_softmax, one block per graph ----------------
__global__ __launch_bounds__(128) void k_fc2_lsm(const float* __restrict__ f1,
                                                 const float* __restrict__ Wc2,
                                                 const float* __restrict__ bc2,
                                                 float* __restrict__ out) {
  int g = blockIdx.x;
  __shared__ float row[D];
  __shared__ float logit[CC];
  row[threadIdx.x] = f1[(long)g * D + threadIdx.x];
  __syncthreads();
  if (threadIdx.x < CC) {
    float s = bc2[threadIdx.x];
    for (int k = 0; k < D; k++) s += row[k] * Wc2[k * CC + threadIdx.x];
    logit[threadIdx.x] = s;
  }
  __syncthreads();
  if (threadIdx.x == 0) {
    float m = logit[0];
    for (int c = 1; c < CC; c++) m = fmaxf(m, logit[c]);
    float se = 0.f;
    for (int c = 0; c < CC; c++) se += __expf(logit[c] - m);
    float lse = m + __logf(se);
    for (int c = 0; c < CC; c++) out[(long)g * CC + c] = logit[c] - lse;
  }
}

extern "C" void kernel_launch(void* const* d_in, const int* in_sizes, int n_in,
                              void* d_out, int out_size, void* d_ws, size_t ws_size,
                              hipStream_t stream) {
  const float* x     = (const float*)d_in[0];
  const int*   ei    = (const int*)d_in[1];     // [2,E]: src then dst
  const int*   batch = (const int*)d_in[2];
  const float* W1    = (const float*)d_in[3];   // [B,128,128]
  const float* b1    = (const float*)d_in[4];   // [B,128]
  const float* g1    = (const float*)d_in[5];
  const float* be1   = (const float*)d_in[6];
  const float* W2    = (const float*)d_in[7];
  const float* b2    = (const float*)d_in[8];
  const float* g2    = (const float*)d_in[9];
  const float* be2   = (const float*)d_in[10];
  const float* bn_g  = (const float*)d_in[11];  // [640]
  const float* bn_b  = (const float*)d_in[12];
  const float* Wc1   = (const float*)d_in[13];  // [640,128]
  const float* bc1   = (const float*)d_in[14];
  const float* Wc2   = (const float*)d_in[15];  // [128,10]
  const float* bc2   = (const float*)d_in[16];

  int N = in_sizes[0] / D;
  int E = in_sizes[1] / 2;
  long ND = (long)N * D;

  // workspace layout
  float* S0    = (float*)d_ws;
  float* S1    = S0 + ND;
  float* stats = S1 + ND;                 // 2*128
  float* feats = stats + 2 * D;           // 512*640
  float* csums = feats + (long)GG * BD;   // 2*640
  float* fbn   = csums + 2 * BD;          // 512*640
  float* f1    = fbn + (long)GG * BD;     // 512*128

  hipMemsetAsync(feats, 0, (size_t)GG * BD * sizeof(float), stream);

  int gE = (int)((ND + 255) / 256);                    // elementwise grid
  int gS = (int)(((long)E * 32 + 255) / 256);          // edge scatter grid
  int gG = ((N >> 4) + 7) / 8;                         // gemm grid (waves of 16 rows)
  int gP = (int)(((long)N * 32 + 255) / 256);          // pool grid

  const float* xin = x;
  for (int i = 0; i < BBLK; i++) {
    float* A = (i & 1) ? S1 : S0;   // aggregation buffer, also block output
    float* H = (i & 1) ? S0 : S1;   // intermediate hidden

    k_scale<<<gE, 256, 0, stream>>>(xin, A, ND);
    k_edge_scatter<<<gS, 256, 0, stream>>>(xin, ei, ei + E, A, E);

    k_gemm_relu<<<gG, 256, 0, stream>>>(A, W1 + (long)i * D * D, b1 + i * D, H, N);
    hipMemsetAsync(stats, 0, 2 * D * sizeof(float), stream);
    k_colstats128<<<256, 256, 0, stream>>>(H, N, stats);
    k_bn<<<gE, 256, 0, stream>>>(H, H, stats, g1 + i * D, be1 + i * D, N, D);

    k_gemm_relu<<<gG, 256, 0, stream>>>(H, W2 + (long)i * D * D, b2 + i * D, A, N);
    hipMemsetAsync(stats, 0, 2 * D * sizeof(float), stream);
    k_colstats128<<<256, 256, 0, stream>>>(A, N, stats);
    k_bn<<<gE, 256, 0, stream>>>(A, A, stats, g2 + i * D, be2 + i * D, N, D);

    k_pool<<<gP, 256, 0, stream>>>(A, batch, feats, N, i * D);
    xin = A;
  }

  // classifier head
  k_colstats_small<<<(BD + 255) / 256, 256, 0, stream>>>(feats, GG, BD, csums);
  k_bn<<<(int)(((long)GG * BD + 255) / 256), 256, 0, stream>>>(feats, fbn, csums,
                                                               bn_g, bn_b, GG, BD);
  k_fc1<<<(GG * D + 255) / 256, 256, 0, stream>>>(fbn, Wc1, bc1, f1);
  k_fc2_lsm<<<GG, 128, 0, stream>>>(f1, Wc2, bc2, (float*)d_out);
}